// _YZ_74758200754818
// MI455X (gfx1250) — compile-verified
//
#include <hip/hip_runtime.h>
#include <stdint.h>

#define AS3 __attribute__((address_space(3)))
#define AS1 __attribute__((address_space(1)))

typedef int v4i __attribute__((vector_size(16)));   // matches builtin's expected pointee

#if defined(__HIP_DEVICE_COMPILE__) && __has_builtin(__builtin_amdgcn_global_load_async_to_lds_b128)
#define USE_ASYNC_LDS 1
#else
#define USE_ASYNC_LDS 0
#endif

// Fused: (dwconv3 + dwconv5 along H) -> BN affine -> 2x2 avgpool -> ReLU
// One block per (b,c) plane. Plane staged in LDS with 2 ghost rows each side.
__launch_bounds__(256, 4)
__global__ void fused_dwconv_bn_pool_relu(
    const float* __restrict__ x,
    const float* __restrict__ w3,
    const float* __restrict__ w5,
    const float* __restrict__ gamma,
    const float* __restrict__ beta,
    const float* __restrict__ rmean,
    const float* __restrict__ rvar,
    float* __restrict__ out)
{
    constexpr int C = 192, H = 56, W = 56, HO = 28, WO = 28;
    constexpr int PLANE = H * W;                 // 3136 floats = 12544 B
    constexpr int LROWS = H + 4;                 // 2 ghost rows top + bottom
    __shared__ __align__(16) float tile[LROWS * W];   // 3360 floats = 13440 B

    const int plane = blockIdx.x;                // 0 .. B*C-1
    const int c     = plane % C;
    const int tid   = threadIdx.x;

    // ---- per-channel constants (uniform -> scalar path) ----
    const float wc0 = w5[c * 5 + 0];
    const float wc1 = w5[c * 5 + 1] + w3[c * 3 + 0];
    const float wc2 = w5[c * 5 + 2] + w3[c * 3 + 1];
    const float wc3 = w5[c * 5 + 3] + w3[c * 3 + 2];
    const float wc4 = w5[c * 5 + 4];
    // y_top + y_bot combined 6-tap weights
    const float e0 = wc0;
    const float e1 = wc1 + wc0;
    const float e2 = wc2 + wc1;
    const float e3 = wc3 + wc2;
    const float e4 = wc4 + wc3;
    const float e5 = wc4;
    const float scale  = gamma[c] * rsqrtf(rvar[c] + 1e-5f);
    const float shift  = beta[c] - rmean[c] * scale;
    const float qscale = 0.25f * scale;

    // ---- zero ghost rows (224 floats total) ----
    if (tid < 2 * W) {
        tile[tid] = 0.0f;                        // rows -2,-1
        tile[(H + 2) * W + tid] = 0.0f;          // rows 56,57
    }

    // ---- stage plane into LDS (contiguous, 128-bit chunks) ----
    const float4* gsrc = (const float4*)(x + (size_t)plane * PLANE);
    float* ldst = tile + 2 * W;
    for (int chunk = tid; chunk < PLANE / 4; chunk += 256) {
#if USE_ASYNC_LDS
        // AS3 pointers are 32-bit; low 32 bits of a generic shared pointer
        // are the LDS byte offset, so route through an integer cast.
        __builtin_amdgcn_global_load_async_to_lds_b128(
            (AS1 v4i*)(uint64_t)(uintptr_t)(gsrc + chunk),
            (AS3 v4i*)(uint32_t)(uintptr_t)(ldst + 4 * chunk),
            0 /*imm offset*/, 0 /*cpol*/);
#else
        float4 v = gsrc[chunk];
        *(float4*)(ldst + 4 * chunk) = v;
#endif
    }

#if USE_ASYNC_LDS
  #if __has_builtin(__builtin_amdgcn_s_wait_asynccnt)
    __builtin_amdgcn_s_wait_asynccnt(0);
  #else
    asm volatile("s_wait_asynccnt 0" ::: "memory");
  #endif
#endif
    __syncthreads();

    // ---- compute 28x28 pooled outputs from LDS ----
    float* obase = out + (size_t)plane * (HO * WO);
    const float2* t2 = (const float2*)tile;      // float2 = one column pair
    for (int p = tid; p < HO * WO; p += 256) {
        const int ho = p / WO;
        const int wo = p - ho * WO;
        const int base = (2 * ho) * (W / 2) + wo;   // padded row 2*ho, col pair wo
        float2 r0 = t2[base];
        float2 r1 = t2[base + 1 * (W / 2)];
        float2 r2 = t2[base + 2 * (W / 2)];
        float2 r3 = t2[base + 3 * (W / 2)];
        float2 r4 = t2[base + 4 * (W / 2)];
        float2 r5 = t2[base + 5 * (W / 2)];
        float s = e0 * (r0.x + r0.y);
        s = fmaf(e1, r1.x + r1.y, s);
        s = fmaf(e2, r2.x + r2.y, s);
        s = fmaf(e3, r3.x + r3.y, s);
        s = fmaf(e4, r4.x + r4.y, s);
        s = fmaf(e5, r5.x + r5.y, s);
        float o = fmaf(s, qscale, shift);
        obase[p] = o > 0.0f ? o : 0.0f;
    }
}

extern "C" void kernel_launch(void* const* d_in, const int* in_sizes, int n_in,
                              void* d_out, int out_size, void* d_ws, size_t ws_size,
                              hipStream_t stream) {
    (void)in_sizes; (void)n_in; (void)out_size; (void)d_ws; (void)ws_size;
    const float* x     = (const float*)d_in[0];
    const float* w3    = (const float*)d_in[1];
    const float* w5    = (const float*)d_in[2];
    const float* gamma = (const float*)d_in[3];
    const float* beta  = (const float*)d_in[4];
    const float* rmean = (const float*)d_in[5];
    const float* rvar  = (const float*)d_in[6];

    const int B = 32, C = 192;
    dim3 grid(B * C);
    dim3 block(256);
    fused_dwconv_bn_pool_relu<<<grid, block, 0, stream>>>(
        x, w3, w5, gamma, beta, rmean, rvar, (float*)d_out);
}